// Context_8383776162380
// MI455X (gfx1250) — compile-verified
//
#include <hip/hip_runtime.h>
#include <hip/hip_bf16.h>

typedef __attribute__((ext_vector_type(2))) float v2f;
typedef __attribute__((ext_vector_type(4))) float vf4;
typedef __attribute__((ext_vector_type(8))) float v8f;

#define SEGS 64
#define DIM  128
#define CHUNK 2048   // rows per block in pass 1

// ---------------------------------------------------------------------------
// Pass 1: segment sum + counts. batch_id is sorted, so each wave keeps a
// running per-lane float4 accumulator and only touches global atomics when the
// segment id changes (rare: segment length ~15625 rows). h_V is streamed with
// non-temporal loads (512 MB, zero reuse).
// ---------------------------------------------------------------------------
__global__ __launch_bounds__(256) void seg_sum_kernel(
    const float* __restrict__ hV, const int* __restrict__ bid,
    float* __restrict__ seg_sum, float* __restrict__ seg_cnt, int N)
{
    const int lane = threadIdx.x & 31;
    const int wave = threadIdx.x >> 5;          // 8 waves per block
    int rBeg = blockIdx.x * CHUNK + wave;
    int rEnd = blockIdx.x * CHUNK + CHUNK;
    if (rEnd > N) rEnd = N;

    vf4 acc = (vf4)0.0f;
    float cnt = 0.0f;
    int cur = -1;

    for (int r = rBeg; r < rEnd; r += 8) {
        int s = bid[r];                          // wave-uniform
        if (s != cur) {
            if (cur >= 0 && cnt > 0.0f) {
                float* base = seg_sum + (size_t)cur * DIM + lane * 4;
                atomicAdd(base + 0, acc.x);
                atomicAdd(base + 1, acc.y);
                atomicAdd(base + 2, acc.z);
                atomicAdd(base + 3, acc.w);
                if (lane == 0) atomicAdd(seg_cnt + cur, cnt);
            }
            cur = s; acc = (vf4)0.0f; cnt = 0.0f;
        }
        const vf4* rowp = (const vf4*)(hV + (size_t)r * DIM);
        vf4 v = __builtin_nontemporal_load(rowp + lane);  // 32 lanes x 16B = full row
        acc += v;
        cnt += 1.0f;
    }
    if (cur >= 0 && cnt > 0.0f) {
        float* base = seg_sum + (size_t)cur * DIM + lane * 4;
        atomicAdd(base + 0, acc.x);
        atomicAdd(base + 1, acc.y);
        atomicAdd(base + 2, acc.z);
        atomicAdd(base + 3, acc.w);
        if (lane == 0) atomicAdd(seg_cnt + cur, cnt);
    }
}

// ---------------------------------------------------------------------------
// Pass 2: c_V = seg_sum/max(cnt,1); g = sigmoid(relu(c_V@W1+b1)@W2+b2).
// Single 1024-thread block (32 waves). Each wave owns one 16x16 output tile
// (mt in 0..3, nt in 0..7) and runs 32 V_WMMA_F32_16X16X4_F32 accumulation
// steps per layer. A/B fragment layout per ISA 7.12.2:
//   A 16x4 f32: lanes 0-15 hold {K=k0,k0+1}, lanes 16-31 hold {K=k0+2,k0+3}
//   B  4x16 f32: same K split, N striped across lanes 0-15 / 16-31
//   C/D 16x16 f32: VGPR r -> row r (lanes 0-15) / row r+8 (lanes 16-31)
// LDS buffer is reused in place for the hidden activations (barriered).
// ---------------------------------------------------------------------------
__global__ __launch_bounds__(1024) void mlp_gate_kernel(
    const float* __restrict__ W1, const float* __restrict__ b1,
    const float* __restrict__ W2, const float* __restrict__ b2,
    const float* __restrict__ seg_sum, const float* __restrict__ seg_cnt,
    float* __restrict__ g)
{
    __shared__ float Abuf[SEGS * DIM];           // 32 KB

    const int tid = threadIdx.x;
    // c_V into LDS
    for (int i = tid; i < SEGS * DIM; i += 1024) {
        float c = seg_cnt[i >> 7];
        c = (c < 1.0f) ? 1.0f : c;
        Abuf[i] = seg_sum[i] / c;
    }
    __syncthreads();

    const int wv   = tid >> 5;
    const int lane = tid & 31;
    const int mt   = wv >> 3;                    // 0..3  (M tile)
    const int nt   = wv & 7;                     // 0..7  (N tile)
    const bool hi  = (lane >= 16);
    const int l15  = lane & 15;
    const int n    = nt * 16 + l15;              // output column for this lane
    const int arow = mt * 16 + l15;              // A row for this lane

    // ---- layer 1: H = relu(c_V @ W1 + b1) ----
    v8f acc = {};
    for (int k0 = 0; k0 < DIM; k0 += 4) {
        int ka = k0 + (hi ? 2 : 0);
        const float* ap = Abuf + arow * DIM + ka;
        v2f a; a.x = ap[0]; a.y = ap[1];
        v2f b; b.x = W1[ka * DIM + n]; b.y = W1[(ka + 1) * DIM + n];
        acc = __builtin_amdgcn_wmma_f32_16x16x4_f32(
                  false, a, false, b, (short)0, acc, false, false);
    }
    float bias1 = b1[n];
    float h[8];
#pragma unroll
    for (int r = 0; r < 8; ++r) {
        float t = acc[r] + bias1;
        h[r] = (t > 0.0f) ? t : 0.0f;
    }
    __syncthreads();                             // all layer-1 reads of Abuf done
#pragma unroll
    for (int r = 0; r < 8; ++r) {
        int m = mt * 16 + r + (hi ? 8 : 0);
        Abuf[m * DIM + n] = h[r];                // hidden acts reuse LDS in place
    }
    __syncthreads();

    // ---- layer 2: g = sigmoid(H @ W2 + b2) ----
    v8f acc2 = {};
    for (int k0 = 0; k0 < DIM; k0 += 4) {
        int ka = k0 + (hi ? 2 : 0);
        const float* ap = Abuf + arow * DIM + ka;
        v2f a; a.x = ap[0]; a.y = ap[1];
        v2f b; b.x = W2[ka * DIM + n]; b.y = W2[(ka + 1) * DIM + n];
        acc2 = __builtin_amdgcn_wmma_f32_16x16x4_f32(
                   false, a, false, b, (short)0, acc2, false, false);
    }
    float bias2 = b2[n];
#pragma unroll
    for (int r = 0; r < 8; ++r) {
        int m = mt * 16 + r + (hi ? 8 : 0);
        float t = acc2[r] + bias2;
        g[m * DIM + n] = 1.0f / (1.0f + __expf(-t));
    }
}

// ---------------------------------------------------------------------------
// Pass 3: out = h_V * g[batch_id]. One float4 per thread; 32 threads per row,
// so batch_id is wave-uniform and g rows stay hot in L2 (g is only 32 KB).
// Non-temporal streaming for the 512 MB in / 512 MB out.
// ---------------------------------------------------------------------------
__global__ __launch_bounds__(256) void modulate_kernel(
    const float* __restrict__ hV, const int* __restrict__ bid,
    const float* __restrict__ g, float* __restrict__ out, long total4)
{
    long i = (long)blockIdx.x * blockDim.x + threadIdx.x;
    if (i >= total4) return;
    long row = i >> 5;
    int  c4  = (int)(i & 31);
    int  s   = bid[row];
    vf4 gv = *(const vf4*)(g + (size_t)s * DIM + c4 * 4);
    vf4 hv = __builtin_nontemporal_load((const vf4*)hV + i);
    __builtin_nontemporal_store(hv * gv, (vf4*)out + i);
}

// ---------------------------------------------------------------------------
extern "C" void kernel_launch(void* const* d_in, const int* in_sizes, int n_in,
                              void* d_out, int out_size, void* d_ws, size_t ws_size,
                              hipStream_t stream)
{
    const float* hV  = (const float*)d_in[0];
    const int*   bid = (const int*)  d_in[1];
    const float* W1  = (const float*)d_in[2];
    const float* b1  = (const float*)d_in[3];
    const float* W2  = (const float*)d_in[4];
    const float* b2  = (const float*)d_in[5];
    float* out = (float*)d_out;
    const int N = in_sizes[1];                   // batch_id length

    float* ws      = (float*)d_ws;
    float* seg_sum = ws;                          // 64*128
    float* seg_cnt = ws + SEGS * DIM;             // 64
    float* g       = ws + SEGS * DIM + SEGS;      // 64*128

    // zero the accumulators every call (deterministic, graph-capture safe)
    hipMemsetAsync(ws, 0, (SEGS * DIM + SEGS) * sizeof(float), stream);

    int blocks1 = (N + CHUNK - 1) / CHUNK;
    seg_sum_kernel<<<blocks1, 256, 0, stream>>>(hV, bid, seg_sum, seg_cnt, N);

    mlp_gate_kernel<<<1, 1024, 0, stream>>>(W1, b1, W2, b2, seg_sum, seg_cnt, g);

    long total4 = (long)N * (DIM / 4);
    int blocks3 = (int)((total4 + 255) / 256);
    modulate_kernel<<<blocks3, 256, 0, stream>>>(hV, bid, g, out, total4);
}